// Router_2302102471519
// MI455X (gfx1250) — compile-verified
//
#include <hip/hip_runtime.h>

typedef __attribute__((ext_vector_type(2))) float v2f;
typedef __attribute__((ext_vector_type(8))) float v8f;

#define NTOK 16384
#define HDIM 2048
#define NEXP 64
#define SINKHORN_ITERS 40

// ---------------------------------------------------------------------------
// Kernel 1: fused jitter + router GEMM (FP32 WMMA 16x16x4) -> cost = exp(logits)
// One wave = 16 tokens x 64 experts. 8 waves per block -> 128 tokens / block.
// A layout (16x4 f32): lanes 0-15 hold K=0,1 ; lanes 16-31 hold K=2,3.
// B layout (4x16 f32): mirrored; N = lane&15, same K split.
// C/D (16x16 f32): VGPR r -> M=r (lanes 0-15) / M=r+8 (lanes 16-31), N=lane&15.
// ---------------------------------------------------------------------------
__global__ __launch_bounds__(256)
void router_gemm_cost_kernel(const float* __restrict__ x,
                             const float* __restrict__ noise,
                             const float* __restrict__ W,
                             float* __restrict__ cost) {
    const int lane = threadIdx.x & 31;
    const int wave = threadIdx.x >> 5;
    const int tile = blockIdx.x * 8 + wave;        // 1024 M-tiles total
    const int r16  = lane & 15;
    const int hi   = lane >> 4;                    // 0 -> K{0,1}, 1 -> K{2,3}

    const long row = (long)tile * 16 + r16;
    const float* xr = x     + row * HDIM;
    const float* nr = noise + row * HDIM;
    const float* w0 = W + (long)r16 * HDIM;        // expert r16 (+16/32/48 below)

    v8f acc0 = {}, acc1 = {}, acc2 = {}, acc3 = {};

    #pragma unroll 4
    for (int k0 = 0; k0 < HDIM; k0 += 4) {
        const int kk = k0 + hi * 2;
        v2f xv = *(const v2f*)(xr + kk);
        v2f nv = *(const v2f*)(nr + kk);
        v2f a  = xv * (0.99f + 0.02f * nv);        // training jitter fused

        v2f b0 = *(const v2f*)(w0 +  0 * HDIM + kk);
        v2f b1 = *(const v2f*)(w0 + 16 * HDIM + kk);
        v2f b2 = *(const v2f*)(w0 + 32 * HDIM + kk);
        v2f b3 = *(const v2f*)(w0 + 48 * HDIM + kk);

        acc0 = __builtin_amdgcn_wmma_f32_16x16x4_f32(false, a, false, b0, (short)0, acc0, false, false);
        acc1 = __builtin_amdgcn_wmma_f32_16x16x4_f32(false, a, false, b1, (short)0, acc1, false, false);
        acc2 = __builtin_amdgcn_wmma_f32_16x16x4_f32(false, a, false, b2, (short)0, acc2, false, false);
        acc3 = __builtin_amdgcn_wmma_f32_16x16x4_f32(false, a, false, b3, (short)0, acc3, false, false);
    }

    const long base_row = (long)tile * 16 + (hi ? 8 : 0);
    #pragma unroll
    for (int r = 0; r < 8; ++r) {
        long o = (base_row + r) * NEXP + r16;
        cost[o +  0] = __expf(acc0[r]);
        cost[o + 16] = __expf(acc1[r]);
        cost[o + 32] = __expf(acc2[r]);
        cost[o + 48] = __expf(acc3[r]);
    }
}

// ---------------------------------------------------------------------------
// Kernel 2: scores[i,:] = cost[i,:] / rowsum(cost[i,:])  (== softmax(logits))
// ---------------------------------------------------------------------------
__global__ __launch_bounds__(256)
void softmax_scores_kernel(const float* __restrict__ cost,
                           float* __restrict__ scores) {
    const long i = (long)blockIdx.x * blockDim.x + threadIdx.x;
    if (i >= NTOK) return;
    const float4* cr = (const float4*)(cost + i * NEXP);
    float4 v[16];
    float s = 0.f;
    #pragma unroll
    for (int q = 0; q < 16; ++q) {
        v[q] = cr[q];
        s += v[q].x + v[q].y + v[q].z + v[q].w;
    }
    const float inv = 1.f / s;
    float4* out = (float4*)(scores + i * NEXP);
    #pragma unroll
    for (int q = 0; q < 16; ++q) {
        float4 o = v[q];
        o.x *= inv; o.y *= inv; o.z *= inv; o.w *= inv;
        out[q] = o;
    }
}

// ---------------------------------------------------------------------------
// Sinkhorn init: d1 = 1, colacc = 0
// ---------------------------------------------------------------------------
__global__ void sinkhorn_init_kernel(float* __restrict__ d1,
                                     float* __restrict__ colacc) {
    const int e = threadIdx.x;
    if (e < NEXP) { d1[e] = 1.f; colacc[e] = 0.f; }
}

// ---------------------------------------------------------------------------
// Sinkhorn row half-step: d0[i] = (1/N)/(cost[i,:].d1 + eps), and accumulate
// column sums of d0[i]*cost[i,:] into colacc via LDS atomics.
// ---------------------------------------------------------------------------
__global__ __launch_bounds__(256)
void sinkhorn_row_kernel(const float* __restrict__ cost,
                         const float* __restrict__ d1,
                         float* __restrict__ d0,
                         float* __restrict__ colacc) {
    __shared__ float sd1[NEXP];
    __shared__ float scol[NEXP];
    const int t = threadIdx.x;
    if (t < NEXP) { sd1[t] = d1[t]; scol[t] = 0.f; }
    __syncthreads();

    const long i = (long)blockIdx.x * blockDim.x + t;
    const float4* cr = (const float4*)(cost + i * NEXP);
    float dot = 0.f;
    #pragma unroll
    for (int q = 0; q < 16; ++q) {
        float4 v = cr[q];
        dot += v.x * sd1[4 * q + 0] + v.y * sd1[4 * q + 1] +
               v.z * sd1[4 * q + 2] + v.w * sd1[4 * q + 3];
    }
    const float d = (1.f / (float)NTOK) / (dot + 1e-8f);
    d0[i] = d;
    #pragma unroll
    for (int q = 0; q < 16; ++q) {
        float4 v = cr[q];                       // L0 hit on second pass
        atomicAdd(&scol[4 * q + 0], v.x * d);   // ds_add_f32
        atomicAdd(&scol[4 * q + 1], v.y * d);
        atomicAdd(&scol[4 * q + 2], v.z * d);
        atomicAdd(&scol[4 * q + 3], v.w * d);
    }
    __syncthreads();
    if (t < NEXP) atomicAdd(&colacc[t], scol[t]);
}

// ---------------------------------------------------------------------------
// Sinkhorn column half-step: d1[e] = (1/E)/(colacc[e]+eps); reset colacc.
// ---------------------------------------------------------------------------
__global__ void sinkhorn_col_kernel(float* __restrict__ d1,
                                    float* __restrict__ colacc) {
    const int e = threadIdx.x;
    if (e < NEXP) {
        d1[e] = (1.f / (float)NEXP) / (colacc[e] + 1e-8f);
        colacc[e] = 0.f;
    }
}

// ---------------------------------------------------------------------------
// Top-2 over balanced matrix (d0 row-scale cannot change ordering), gather
// expert weights from the plain-softmax scores. Stable tie-break == lax.top_k.
// ---------------------------------------------------------------------------
__global__ __launch_bounds__(256)
void topk_gather_kernel(const float* __restrict__ cost,
                        const float* __restrict__ d1,
                        const float* __restrict__ scores,
                        float* __restrict__ ew,
                        float* __restrict__ idx_out) {
    __shared__ float sd1[NEXP];
    const int t = threadIdx.x;
    if (t < NEXP) sd1[t] = d1[t];
    __syncthreads();

    const long i = (long)blockIdx.x * blockDim.x + t;
    const float* cr = cost + i * NEXP;
    float v0 = -3.402823466e+38f, v1 = -3.402823466e+38f;
    int i0 = 0, i1 = 0;
    for (int e = 0; e < NEXP; ++e) {
        const float w = cr[e] * sd1[e];
        if (w > v0)      { v1 = v0; i1 = i0; v0 = w; i0 = e; }
        else if (w > v1) { v1 = w; i1 = e; }
    }
    const float* sr = scores + i * NEXP;
    ew[i * 2 + 0] = sr[i0];
    ew[i * 2 + 1] = sr[i1];
    idx_out[i * 2 + 0] = (float)i0;
    idx_out[i * 2 + 1] = (float)i1;
}

// ---------------------------------------------------------------------------
// Launch: out layout = [expert_weights 32768][indices 32768][scores 1048576]
// ws layout = [cost 16384*64][d0 16384][d1 64][colacc 64]   (~4.26 MB)
// ---------------------------------------------------------------------------
extern "C" void kernel_launch(void* const* d_in, const int* in_sizes, int n_in,
                              void* d_out, int out_size, void* d_ws, size_t ws_size,
                              hipStream_t stream) {
    const float* x     = (const float*)d_in[0];
    const float* noise = (const float*)d_in[1];
    const float* W     = (const float*)d_in[2];

    float* out     = (float*)d_out;
    float* ew      = out;                       // [16384,2]
    float* idx     = out + NTOK * 2;            // [16384,2] (as float values)
    float* scores  = out + NTOK * 4;            // [16384,64]

    float* cost    = (float*)d_ws;              // exp(logits)
    float* d0      = cost + (long)NTOK * NEXP;
    float* d1      = d0 + NTOK;
    float* colacc  = d1 + NEXP;

    router_gemm_cost_kernel<<<NTOK / 128, 256, 0, stream>>>(x, noise, W, cost);
    softmax_scores_kernel<<<NTOK / 256, 256, 0, stream>>>(cost, scores);
    sinkhorn_init_kernel<<<1, NEXP, 0, stream>>>(d1, colacc);
    for (int it = 0; it < SINKHORN_ITERS; ++it) {
        sinkhorn_row_kernel<<<NTOK / 256, 256, 0, stream>>>(cost, d1, d0, colacc);
        sinkhorn_col_kernel<<<1, NEXP, 0, stream>>>(d1, colacc);
    }
    topk_gather_kernel<<<NTOK / 256, 256, 0, stream>>>(cost, d1, scores, ew, idx);
}